// Encoder_22986664968867
// MI455X (gfx1250) — compile-verified
//
#include <hip/hip_runtime.h>
#include <hip/hip_bf16.h>
#include <math.h>

// ---------------------------------------------------------------------------
// Transformer encoder (12 blocks) for MI455X / gfx1250, full f32 precision.
// All GEMMs + attention use V_WMMA_F32_16X16X4_F32 (wave32 WMMA).
// Global->LDS staging uses GLOBAL_LOAD_ASYNC_TO_LDS_B128 with double buffering
// (ASYNCcnt), falling back to sync loads if the builtin is unavailable.
// ---------------------------------------------------------------------------

typedef float v2f __attribute__((ext_vector_type(2)));
typedef float v8f __attribute__((ext_vector_type(8)));
typedef int   v4i __attribute__((vector_size(16)));               // GCC-style, matches builtin V4i
typedef __attribute__((address_space(1))) v4i* global_v4i_ptr;    // global src
typedef __attribute__((address_space(3))) v4i* lds_v4i_ptr;       // LDS dst

#define NB   12
#define DMOD 512
#define NH   8
#define DH   64
#define DHID 1024
#define BB   2
#define SS   2048
#define MROWS (BB * SS)          // 4096
#define QKV_STRIDE (NH * 3 * DH) // 1536

#if __has_builtin(__builtin_amdgcn_global_load_async_to_lds_b128) && \
    __has_builtin(__builtin_amdgcn_s_wait_asynccnt)
#define USE_ASYNC_LDS 1
#endif

__device__ __forceinline__ void async_copy16(void* lds, const void* g) {
#ifdef USE_ASYNC_LDS
    __builtin_amdgcn_global_load_async_to_lds_b128(
        (global_v4i_ptr)g, (lds_v4i_ptr)lds, 0, 0);
#else
    *(float4*)lds = *(const float4*)g;
#endif
}

__device__ __forceinline__ void async_join() {
#ifdef USE_ASYNC_LDS
    __builtin_amdgcn_s_wait_asynccnt(0);
#endif
}

__device__ __forceinline__ v8f zero8() {
    v8f z = {0.f, 0.f, 0.f, 0.f, 0.f, 0.f, 0.f, 0.f};
    return z;
}

__device__ __forceinline__ v8f wmma_f32(v2f a, v2f b, v8f c) {
    // D = A(16x4) * B(4x16) + C(16x16), full f32
    return __builtin_amdgcn_wmma_f32_16x16x4_f32(
        false, a, false, b, (short)0, c, false, false);
}

// ---------------------------------------------------------------------------
// GEMM: C[M,N] = A[M,K] @ B[K,N]  (+bias) (+relu) (+residual)
// flags bit0 = relu, bit1 = add residual
// block = 128 threads (4 waves); block tile 64x64; K step 16; double-buffered
// async global->LDS staging.
// ---------------------------------------------------------------------------
__global__ __launch_bounds__(128) void gemm_f32_wmma(
    const float* __restrict__ A, const float* __restrict__ Bw,
    const float* __restrict__ bias, const float* __restrict__ Res,
    float* __restrict__ C, int M, int N, int K, int flags)
{
    __shared__ __align__(16) float As[2][64 * 16];
    __shared__ __align__(16) float Bs[2][16 * 64];

    const int t    = threadIdx.x;
    const int wave = t >> 5;
    const int lane = t & 31;
    const int hl   = lane >> 4;   // half of wave (selects K pair)
    const int l16  = lane & 15;
    const int bm   = blockIdx.y * 64;
    const int bn   = blockIdx.x * 64;

    // staging assignments (64x16 A tile, 16x64 B tile; 16 bytes x2 each)
    const int arow = t >> 1, acs = (t & 1) * 8;
    const int brow = t >> 3, bcs = (t & 7) * 8;
    const float* gA = &A[(size_t)(bm + arow) * K + acs];
    const float* gB = &Bw[(size_t)brow * N + bn + bcs];

    auto stage = [&](int k0, int buf) {
        async_copy16(&As[buf][arow * 16 + acs],     gA + k0);
        async_copy16(&As[buf][arow * 16 + acs + 4], gA + k0 + 4);
        async_copy16(&Bs[buf][brow * 64 + bcs],     gB + (size_t)k0 * N);
        async_copy16(&Bs[buf][brow * 64 + bcs + 4], gB + (size_t)k0 * N + 4);
    };

    v8f acc[4];
    #pragma unroll
    for (int i = 0; i < 4; ++i) acc[i] = zero8();

    stage(0, 0);
    async_join();
    __syncthreads();

    int buf = 0;
    for (int k0 = 0; k0 < K; k0 += 16) {
        if (k0 + 16 < K) stage(k0 + 16, buf ^ 1);

        #pragma unroll
        for (int c = 0; c < 4; ++c) {
            // A fragment: lane holds row (l16), K pair (c*4 + 2*hl)
            v2f a = *(const v2f*)&As[buf][(wave * 16 + l16) * 16 + c * 4 + 2 * hl];
            #pragma unroll
            for (int nt = 0; nt < 4; ++nt) {
                v2f b;
                b[0] = Bs[buf][(c * 4 + 2 * hl) * 64 + nt * 16 + l16];
                b[1] = Bs[buf][(c * 4 + 2 * hl + 1) * 64 + nt * 16 + l16];
                acc[nt] = wmma_f32(a, b, acc[nt]);
            }
        }

        async_join();
        __syncthreads();
        buf ^= 1;
    }

    // --- epilogue: bias / relu / residual ---
    #pragma unroll
    for (int nt = 0; nt < 4; ++nt) {
        #pragma unroll
        for (int i = 0; i < 8; ++i) {
            const int m = bm + wave * 16 + i + 8 * hl;
            const int n = bn + nt * 16 + l16;
            float v = acc[nt][i] + bias[n];
            if (flags & 1) v = fmaxf(v, 0.f);
            if (flags & 2) v += Res[(size_t)m * N + n];
            C[(size_t)m * N + n] = v;
        }
    }
}

// ---------------------------------------------------------------------------
// Flash attention.  QKV layout: [B, S, H*192] with per-(s,h) [q(64),k(64),v(64)].
// grid = (B*H, S/64), block = 128 (4 waves); each wave owns 16 query rows.
// K/V streamed in 16-row tiles via double-buffered async LDS staging;
// online softmax; O accumulated in registers.
// ---------------------------------------------------------------------------
__global__ __launch_bounds__(128) void attn_f32_wmma(
    const float* __restrict__ QKV, float* __restrict__ Out)
{
    __shared__ __align__(16) float Klds[2][16 * 64];
    __shared__ __align__(16) float Vlds[2][16 * 64];
    __shared__ __align__(16) float Plds[4][16 * 16];

    const int t    = threadIdx.x;
    const int wave = t >> 5;
    const int lane = t & 31;
    const int hl   = lane >> 4;
    const int l16  = lane & 15;

    const int bh = blockIdx.x;           // 0 .. B*H-1
    const int b  = bh >> 3;              // H = 8
    const int h  = bh & 7;
    const int qbase = (blockIdx.y * 4 + wave) * 16;

    const float scale = 0.044194173824159216f; // 1/sqrt(512) (ref scales by d_model)
    const float* base = QKV + (size_t)b * SS * QKV_STRIDE + h * 3 * DH;

    // staging assignments for 16x64 K and V tiles
    const int r  = t >> 3;
    const int cs = (t & 7) * 8;
    const float* gK = base + DH     + (size_t)r * QKV_STRIDE + cs;
    const float* gV = base + 2 * DH + (size_t)r * QKV_STRIDE + cs;

    auto stage = [&](int kv, int buf) {
        const size_t off = (size_t)kv * QKV_STRIDE;
        async_copy16(&Klds[buf][r * 64 + cs],     gK + off);
        async_copy16(&Klds[buf][r * 64 + cs + 4], gK + off + 4);
        async_copy16(&Vlds[buf][r * 64 + cs],     gV + off);
        async_copy16(&Vlds[buf][r * 64 + cs + 4], gV + off + 4);
    };

    // --- preload Q fragments: 16 rows x 64 dims as 16 A-fragments ---
    v2f qf[16];
    {
        const float* qrow = base + (size_t)(qbase + l16) * QKV_STRIDE;
        #pragma unroll
        for (int c = 0; c < 16; ++c)
            qf[c] = *(const v2f*)&qrow[c * 4 + 2 * hl];
    }

    v8f o[4];
    #pragma unroll
    for (int i = 0; i < 4; ++i) o[i] = zero8();
    float mrow[8], lsum[8];
    #pragma unroll
    for (int i = 0; i < 8; ++i) { mrow[i] = -INFINITY; lsum[i] = 0.f; }

    stage(0, 0);
    async_join();
    __syncthreads();

    int buf = 0;
    for (int kv = 0; kv < SS; kv += 16) {
        if (kv + 16 < SS) stage(kv + 16, buf ^ 1);

        // --- scores: S(16x16) = Q(16x64) . K(16x64)^T via 16 WMMAs ---
        v8f sc = zero8();
        #pragma unroll
        for (int c = 0; c < 16; ++c) {
            v2f bk = *(const v2f*)&Klds[buf][l16 * 64 + c * 4 + 2 * hl];
            sc = wmma_f32(qf[c], bk, sc);
        }

        // --- online softmax; C-layout: component i <-> row (i + 8*hl) ---
        float mt[8];
        #pragma unroll
        for (int i = 0; i < 8; ++i) {
            float v = sc[i] * scale;
            sc[i] = v;
            v = fmaxf(v, __shfl_xor(v, 1, 32));
            v = fmaxf(v, __shfl_xor(v, 2, 32));
            v = fmaxf(v, __shfl_xor(v, 4, 32));
            v = fmaxf(v, __shfl_xor(v, 8, 32));
            mt[i] = v;
        }
        float alpha[8];
        #pragma unroll
        for (int i = 0; i < 8; ++i) {
            float mn = fmaxf(mrow[i], mt[i]);
            alpha[i] = __expf(mrow[i] - mn);
            mrow[i]  = mn;
        }
        float ps[8];
        #pragma unroll
        for (int i = 0; i < 8; ++i) {
            float p = __expf(sc[i] - mrow[i]);
            sc[i] = p;
            float s = p;
            s += __shfl_xor(s, 1, 32);
            s += __shfl_xor(s, 2, 32);
            s += __shfl_xor(s, 4, 32);
            s += __shfl_xor(s, 8, 32);
            ps[i] = s;
        }
        #pragma unroll
        for (int i = 0; i < 8; ++i) lsum[i] = lsum[i] * alpha[i] + ps[i];
        #pragma unroll
        for (int nt = 0; nt < 4; ++nt)
            #pragma unroll
            for (int i = 0; i < 8; ++i) o[nt][i] *= alpha[i];

        // --- relayout P: C-layout -> A-layout via per-wave LDS tile ---
        #pragma unroll
        for (int i = 0; i < 8; ++i)
            Plds[wave][(i + 8 * hl) * 16 + l16] = sc[i];
        // (same-wave LDS ops are in-order; compiler inserts s_wait_dscnt)

        // --- O += P(16x16) . V(16x64) via 16 WMMAs ---
        #pragma unroll
        for (int c = 0; c < 4; ++c) {
            v2f ap = *(const v2f*)&Plds[wave][l16 * 16 + c * 4 + 2 * hl];
            #pragma unroll
            for (int nt = 0; nt < 4; ++nt) {
                v2f bv;
                bv[0] = Vlds[buf][(c * 4 + 2 * hl) * 64 + nt * 16 + l16];
                bv[1] = Vlds[buf][(c * 4 + 2 * hl + 1) * 64 + nt * 16 + l16];
                o[nt] = wmma_f32(ap, bv, o[nt]);
            }
        }

        async_join();
        __syncthreads();
        buf ^= 1;
    }

    // --- normalize and store O to [B,S,D] (col = h*64 + n) ---
    float inv[8];
    #pragma unroll
    for (int i = 0; i < 8; ++i) inv[i] = 1.f / lsum[i];
    #pragma unroll
    for (int nt = 0; nt < 4; ++nt) {
        #pragma unroll
        for (int i = 0; i < 8; ++i) {
            const int m = qbase + i + 8 * hl;
            const int n = nt * 16 + l16;
            Out[((size_t)b * SS + m) * DMOD + h * DH + n] = o[nt][i] * inv[i];
        }
    }
}

// ---------------------------------------------------------------------------
// LayerNorm: one wave per 512-element row.
// ---------------------------------------------------------------------------
__global__ __launch_bounds__(256) void ln_kernel(
    const float* __restrict__ X, const float* __restrict__ g,
    const float* __restrict__ bta, float* __restrict__ Y)
{
    const int wave = threadIdx.x >> 5;
    const int lane = threadIdx.x & 31;
    const int row  = blockIdx.x * 8 + wave;
    const float* x = X + (size_t)row * DMOD;

    float vals[16];
    float s = 0.f;
    #pragma unroll
    for (int j = 0; j < 16; ++j) { vals[j] = x[lane + j * 32]; s += vals[j]; }
    s += __shfl_xor(s, 1, 32);  s += __shfl_xor(s, 2, 32);
    s += __shfl_xor(s, 4, 32);  s += __shfl_xor(s, 8, 32);
    s += __shfl_xor(s, 16, 32);
    const float mean = s * (1.f / DMOD);

    float v = 0.f;
    #pragma unroll
    for (int j = 0; j < 16; ++j) { float d = vals[j] - mean; v += d * d; }
    v += __shfl_xor(v, 1, 32);  v += __shfl_xor(v, 2, 32);
    v += __shfl_xor(v, 4, 32);  v += __shfl_xor(v, 8, 32);
    v += __shfl_xor(v, 16, 32);
    const float rstd = rsqrtf(v * (1.f / DMOD) + 1e-5f);

    #pragma unroll
    for (int j = 0; j < 16; ++j) {
        const int c = lane + j * 32;
        Y[(size_t)row * DMOD + c] = (vals[j] - mean) * rstd * g[c] + bta[c];
    }
}

__global__ __launch_bounds__(256) void copy4_kernel(
    float4* __restrict__ dst, const float4* __restrict__ src)
{
    const int i = blockIdx.x * 256 + threadIdx.x;
    dst[i] = src[i];
}

// ---------------------------------------------------------------------------
extern "C" void kernel_launch(void* const* d_in, const int* in_sizes, int n_in,
                              void* d_out, int out_size, void* d_ws, size_t ws_size,
                              hipStream_t stream) {
    const float* X     = (const float*)d_in[0];
    const float* qkv_w = (const float*)d_in[1];
    const float* qkv_b = (const float*)d_in[2];
    const float* fc_w  = (const float*)d_in[3];
    const float* fc_b  = (const float*)d_in[4];
    const float* ln1_g = (const float*)d_in[5];
    const float* ln1_b = (const float*)d_in[6];
    const float* w1    = (const float*)d_in[7];
    const float* b1    = (const float*)d_in[8];
    const float* w2    = (const float*)d_in[9];
    const float* b2    = (const float*)d_in[10];
    const float* ln2_g = (const float*)d_in[11];
    const float* ln2_b = (const float*)d_in[12];

    float* ws   = (float*)d_ws;
    float* qkv  = ws;                                  // 4096*1536
    float* ao   = qkv + (size_t)MROWS * 3 * DMOD;      // 4096*512 (also reused as x2)
    float* x1   = ao  + (size_t)MROWS * DMOD;          // 4096*512
    float* xn   = x1  + (size_t)MROWS * DMOD;          // 4096*512
    float* hbuf = xn  + (size_t)MROWS * DMOD;          // 4096*1024
    float* xcur = (float*)d_out;                       // running activations

    // xcur = X
    copy4_kernel<<<(MROWS * DMOD) / (4 * 256), 256, 0, stream>>>(
        (float4*)xcur, (const float4*)X);

    for (int p = 0; p < NB; ++p) {
        // 1) QKV projection: [4096,512] @ [512,1536] + b
        gemm_f32_wmma<<<dim3(3 * DMOD / 64, MROWS / 64), 128, 0, stream>>>(
            xcur, qkv_w + (size_t)p * DMOD * 3 * DMOD, qkv_b + (size_t)p * 3 * DMOD,
            nullptr, qkv, MROWS, 3 * DMOD, DMOD, 0);

        // 2) flash attention -> ao [4096,512]
        attn_f32_wmma<<<dim3(BB * NH, SS / 64), 128, 0, stream>>>(qkv, ao);

        // 3) out proj + residual: x1 = xcur + ao @ fc_w + fc_b
        gemm_f32_wmma<<<dim3(DMOD / 64, MROWS / 64), 128, 0, stream>>>(
            ao, fc_w + (size_t)p * DMOD * DMOD, fc_b + (size_t)p * DMOD,
            xcur, x1, MROWS, DMOD, DMOD, 2);

        // 4) xn = LN1(x1)
        ln_kernel<<<MROWS / 8, 256, 0, stream>>>(
            x1, ln1_g + (size_t)p * DMOD, ln1_b + (size_t)p * DMOD, xn);

        // 5) hbuf = relu(xn @ w1 + b1)   [4096,1024]
        gemm_f32_wmma<<<dim3(DHID / 64, MROWS / 64), 128, 0, stream>>>(
            xn, w1 + (size_t)p * DMOD * DHID, b1 + (size_t)p * DHID,
            nullptr, hbuf, MROWS, DHID, DMOD, 1);

        // 6) x2 = x1 + hbuf @ w2 + b2   (write into ao, now free)
        gemm_f32_wmma<<<dim3(DMOD / 64, MROWS / 64), 128, 0, stream>>>(
            hbuf, w2 + (size_t)p * DHID * DMOD, b2 + (size_t)p * DMOD,
            x1, ao, MROWS, DMOD, DHID, 2);

        // 7) xcur = LN2(x2)
        ln_kernel<<<MROWS / 8, 256, 0, stream>>>(
            ao, ln2_g + (size_t)p * DMOD, ln2_b + (size_t)p * DMOD, xcur);
    }
}